// HomoSAGE_39977555591470
// MI455X (gfx1250) — compile-verified
//
#include <hip/hip_runtime.h>

typedef __attribute__((ext_vector_type(2))) float v2f;
typedef __attribute__((ext_vector_type(8))) float v8f;

#define HIDDEN   128
#define NGRAPHS  128

// ---------------------------------------------------------------- utilities
__global__ void zero_f32(float* __restrict__ p, int n) {
  int i = blockIdx.x * blockDim.x + threadIdx.x;
  int stride = gridDim.x * blockDim.x;
  for (; i < n; i += stride) p[i] = 0.0f;
}

__global__ void inv_deg_kernel(const float* __restrict__ deg,
                               float* __restrict__ inv, int n) {
  int i = blockIdx.x * blockDim.x + threadIdx.x;
  if (i < n) inv[i] = 1.0f / fmaxf(deg[i], 1.0f);
}

// ------------------------------------------------- edge scatter (mean aggr)
// one wave32 per edge; each lane moves 4 of the 128 feature dims.
__global__ __launch_bounds__(256)
void scatter_accum(const float* __restrict__ feat,
                   const int* __restrict__ src,
                   const int* __restrict__ dst,
                   float* __restrict__ agg,
                   float* __restrict__ deg,
                   int nEdges, int doDeg) {
  int edge = (int)((blockIdx.x * blockDim.x + threadIdx.x) >> 5);
  int lane = threadIdx.x & 31;
  if (edge >= nEdges) return;
  int s = src[edge];
  int d = dst[edge];
  const float4 v = *(const float4*)(feat + (size_t)s * HIDDEN + lane * 4);
  float* o = agg + (size_t)d * HIDDEN + lane * 4;
  atomicAdd(o + 0, v.x);
  atomicAdd(o + 1, v.y);
  atomicAdd(o + 2, v.z);
  atomicAdd(o + 3, v.w);
  if (doDeg && lane == 0) atomicAdd(deg + d, 1.0f);
}

// --------------------------------------- fused SAGE layer: WMMA f32 GEMM
// out = relu( diag(inv_deg) * (agg @ wl^T) + bias + xin @ wr^T )
// block = one 16-row M tile (M = 3125*16 = 50000 exactly)
// wave (8/block) = one 16-col N tile (8*16 = 128 = HIDDEN)
// Two independent accumulator chains; 1/deg scaling deferred to the epilogue
// (row scaling commutes with the GEMM), so the hot loop is loads + WMMA only.
__global__ __launch_bounds__(256)
void sage_gemm(const float* __restrict__ agg,
               const float* __restrict__ inv_deg,
               const float* __restrict__ xin,
               const float* __restrict__ wl,
               const float* __restrict__ wr,
               const float* __restrict__ bias,
               float* __restrict__ out) {
  const int mtile = blockIdx.x;
  const int wave  = threadIdx.x >> 5;      // N tile id, 0..7
  const int lane  = threadIdx.x & 31;
  const int half  = lane >> 4;             // selects K+2 (A/B) and M+8 (C/D)
  const int l16   = lane & 15;
  const int m = mtile * 16 + l16;          // this lane's A row
  const int n = wave * 16 + l16;           // this lane's B column

  const float* arow = agg + (size_t)m * HIDDEN + 2 * half;
  const float* xrow = xin + (size_t)m * HIDDEN + 2 * half;
  const float* blp  = wl  + (size_t)n * HIDDEN + 2 * half;  // B[k][n] = wl[n][k]
  const float* brp  = wr  + (size_t)n * HIDDEN + 2 * half;

  v8f acc1 = {};   // agg @ wl^T (unscaled)
  v8f acc2 = {};   // xin @ wr^T
#pragma unroll
  for (int k = 0; k < HIDDEN; k += 4) {
    v2f a1 = *(const v2f*)(arow + k);
    v2f b1 = *(const v2f*)(blp + k);
    v2f a2 = *(const v2f*)(xrow + k);
    v2f b2 = *(const v2f*)(brp + k);
    acc1 = __builtin_amdgcn_wmma_f32_16x16x4_f32(false, a1, false, b1,
                                                 (short)0, acc1, false, false);
    acc2 = __builtin_amdgcn_wmma_f32_16x16x4_f32(false, a2, false, b2,
                                                 (short)0, acc2, false, false);
  }

  // epilogue: per-row 1/deg scale of term 1, bias, relu
  const float bv = bias[n];
  const float* invp = inv_deg + mtile * 16 + 8 * half;  // rows j+8*half, j=0..7
  const float4 i0 = *(const float4*)(invp);
  const float4 i1 = *(const float4*)(invp + 4);
  float invr[8] = {i0.x, i0.y, i0.z, i0.w, i1.x, i1.y, i1.z, i1.w};
#pragma unroll
  for (int j = 0; j < 8; ++j) {
    int row = mtile * 16 + j + 8 * half;   // C/D layout: VGPR j -> M=j or j+8
    float v = fmaf(acc1[j], invr[j], acc2[j] + bv);
    v = fmaxf(v, 0.0f);
    out[(size_t)row * HIDDEN + n] = v;
  }
}

// ---------------------------------------------------------- global mean pool
__global__ __launch_bounds__(256)
void pool_kernel(const float* __restrict__ h,
                 const int* __restrict__ batch,
                 float* __restrict__ gsum,
                 float* __restrict__ gcnt,
                 int nNodes) {
  long long idx = (long long)blockIdx.x * blockDim.x + threadIdx.x;
  int node = (int)(idx >> 7);
  int dim  = (int)(idx & 127);
  if (node >= nNodes) return;
  int g = batch[node];
  atomicAdd(&gsum[g * HIDDEN + dim], h[(size_t)node * HIDDEN + dim]);
  if (dim == 0) atomicAdd(&gcnt[g], 1.0f);
}

// ----------------------------------------- classifier: (gsum/cnt)@cw^T + cb
__global__ __launch_bounds__(256)
void classifier_gemm(const float* __restrict__ gsum,
                     const float* __restrict__ gcnt,
                     const float* __restrict__ cw,
                     const float* __restrict__ cb,
                     float* __restrict__ out) {
  const int mtile = blockIdx.x;            // 8 tiles of 16 graphs
  const int wave  = threadIdx.x >> 5;      // 8 N tiles
  const int lane  = threadIdx.x & 31;
  const int half  = lane >> 4;
  const int l16   = lane & 15;
  const int m = mtile * 16 + l16;
  const int n = wave * 16 + l16;

  const float* arow = gsum + (size_t)m * HIDDEN + 2 * half;
  const float* brow = cw   + (size_t)n * HIDDEN + 2 * half;

  v8f acc = {};
#pragma unroll
  for (int k = 0; k < HIDDEN; k += 4) {
    v2f a = *(const v2f*)(arow + k);
    v2f b = *(const v2f*)(brow + k);
    acc = __builtin_amdgcn_wmma_f32_16x16x4_f32(false, a, false, b,
                                                (short)0, acc, false, false);
  }

  // epilogue: per-row 1/count scale + bias
  const float bv = cb[n];
  const float* cntp = gcnt + mtile * 16 + 8 * half;
  const float4 c0 = *(const float4*)(cntp);
  const float4 c1 = *(const float4*)(cntp + 4);
  float cnt[8] = {c0.x, c0.y, c0.z, c0.w, c1.x, c1.y, c1.z, c1.w};
#pragma unroll
  for (int j = 0; j < 8; ++j) {
    int row = mtile * 16 + j + 8 * half;
    float inv = 1.0f / fmaxf(cnt[j], 1.0f);
    out[(size_t)row * HIDDEN + n] = fmaf(acc[j], inv, bv);
  }
}

// ------------------------------------------------------------------ launch
extern "C" void kernel_launch(void* const* d_in, const int* in_sizes, int n_in,
                              void* d_out, int out_size, void* d_ws, size_t ws_size,
                              hipStream_t stream) {
  const float* x    = (const float*)d_in[0];
  const int*   eidx = (const int*)d_in[1];     // [2, E] flattened
  const int*   batch= (const int*)d_in[2];
  const float* wl1  = (const float*)d_in[3];
  const float* b1   = (const float*)d_in[4];
  const float* wr1  = (const float*)d_in[5];
  const float* wl2  = (const float*)d_in[6];
  const float* b2   = (const float*)d_in[7];
  const float* wr2  = (const float*)d_in[8];
  const float* cw   = (const float*)d_in[9];
  const float* cb   = (const float*)d_in[10];
  float* out = (float*)d_out;

  const int nNodes = in_sizes[0] / HIDDEN;     // 50000
  const int nEdges = in_sizes[1] / 2;          // 800000
  const int featN  = nNodes * HIDDEN;          // 6,400,000

  const int* src = eidx;
  const int* dst = eidx + nEdges;

  // workspace layout (floats)
  float* ws   = (float*)d_ws;
  float* agg  = ws;                    // featN
  float* h1   = agg  + featN;          // featN
  float* h2   = h1   + featN;          // featN
  float* deg  = h2   + featN;          // nNodes
  float* inv  = deg  + nNodes;         // nNodes
  float* gsum = inv  + nNodes;         // 128*128
  float* gcnt = gsum + NGRAPHS*HIDDEN; // 128  (contiguous with gsum for zeroing)

  const int mtiles = nNodes / 16;                    // 3125
  const int scatterBlocks = (nEdges + 7) / 8;        // 8 waves/block, 1 edge/wave
  const int poolBlocks = (nNodes * HIDDEN + 255) / 256;

  // ---- layer 1 aggregation
  zero_f32<<<2048, 256, 0, stream>>>(agg, featN);
  zero_f32<<<(nNodes + 255) / 256, 256, 0, stream>>>(deg, nNodes);
  scatter_accum<<<scatterBlocks, 256, 0, stream>>>(x, src, dst, agg, deg, nEdges, 1);
  inv_deg_kernel<<<(nNodes + 255) / 256, 256, 0, stream>>>(deg, inv, nNodes);
  // ---- layer 1 fused GEMM + bias + relu
  sage_gemm<<<mtiles, 256, 0, stream>>>(agg, inv, x, wl1, wr1, b1, h1);

  // ---- layer 2 aggregation (degree unchanged)
  zero_f32<<<2048, 256, 0, stream>>>(agg, featN);
  scatter_accum<<<scatterBlocks, 256, 0, stream>>>(h1, src, dst, agg, deg, nEdges, 0);
  // ---- layer 2 fused GEMM + bias + relu
  sage_gemm<<<mtiles, 256, 0, stream>>>(agg, inv, h1, wl2, wr2, b2, h2);

  // ---- global mean pool + classifier
  zero_f32<<<((NGRAPHS*HIDDEN + NGRAPHS) + 255) / 256, 256, 0, stream>>>(gsum, NGRAPHS*HIDDEN + NGRAPHS);
  pool_kernel<<<poolBlocks, 256, 0, stream>>>(h2, batch, gsum, gcnt, nNodes);
  classifier_gemm<<<NGRAPHS / 16, 256, 0, stream>>>(gsum, gcnt, cw, cb, out);
}